// TransducerLoss_30794915512814
// MI455X (gfx1250) — compile-verified
//
#include <hip/hip_runtime.h>
#include <hip/hip_bf16.h>
#include <math.h>

#define NEG_INF (-__builtin_inff())

// Fixed problem shape from the reference setup.
#define BB 8
#define TT 256
#define UU 97
#define VV 512
#define BT (BB*TT)   // 2048

typedef __attribute__((ext_vector_type(2))) float v2f;
typedef __attribute__((ext_vector_type(8))) float v8f;

// ---------------------------------------------------------------------------
// Kernel 1: per-(b,t,u) row logsumexp over V=512 + gather blank/label logits.
// One wave per row; 32 lanes x 16 floats, fully coalesced b128 loads.
// Bandwidth-bound: reads 407MB once.
// ---------------------------------------------------------------------------
__global__ void k_row_lse(const float* __restrict__ x,
                          const int*   __restrict__ label,
                          const int*   __restrict__ pblank,
                          float* __restrict__ blank_lp,   // [BT][97]
                          float* __restrict__ lab_lp) {   // [BT][96]
  const int bt   = blockIdx.x;                // 0..2047  (= b*T + t)
  const int w    = threadIdx.x >> 5;
  const int lane = threadIdx.x & 31;
  const int u    = blockIdx.y * 8 + w;
  if (u >= UU) return;

  const int bidx = pblank[0];
  const int b    = bt >> 8;                   // T == 256
  const int lidx = (u < UU - 1) ? label[b * (UU - 1) + u] : -1;

  const float4* px = (const float4*)(x + ((size_t)bt * UU + u) * VV);
  // coalesced: instruction i reads float4 index (i*32 + lane)
  float4 f0 = px[0 * 32 + lane];
  float4 f1 = px[1 * 32 + lane];
  float4 f2 = px[2 * 32 + lane];
  float4 f3 = px[3 * 32 + lane];

  float vals[16] = {f0.x, f0.y, f0.z, f0.w, f1.x, f1.y, f1.z, f1.w,
                    f2.x, f2.y, f2.z, f2.w, f3.x, f3.y, f3.z, f3.w};

  // lane max
  float m = vals[0];
#pragma unroll
  for (int i = 1; i < 16; ++i) m = fmaxf(m, vals[i]);
  // wave32 max reduce
#pragma unroll
  for (int off = 16; off >= 1; off >>= 1) m = fmaxf(m, __shfl_xor(m, off, 32));

  // lane sum of exp(v - m)
  float s = 0.f;
#pragma unroll
  for (int i = 0; i < 16; ++i) s += __expf(vals[i] - m);
#pragma unroll
  for (int off = 16; off >= 1; off >>= 1) s += __shfl_xor(s, off, 32);

  const float lse = m + __logf(s);

  // gather x[blank_idx] and x[label]; element i of this lane is global index
  // g = (i/4)*128 + lane*4 + (i%4)
  float xb = NEG_INF, xl = NEG_INF;
#pragma unroll
  for (int i = 0; i < 16; ++i) {
    const int g = (i >> 2) * 128 + (lane << 2) + (i & 3);
    xb = (g == bidx) ? vals[i] : xb;
    xl = (g == lidx) ? vals[i] : xl;
  }
#pragma unroll
  for (int off = 16; off >= 1; off >>= 1) {
    xb = fmaxf(xb, __shfl_xor(xb, off, 32));
    xl = fmaxf(xl, __shfl_xor(xl, off, 32));
  }

  if (lane == 0) {
    blank_lp[(size_t)bt * UU + u] = xb - lse;
    if (u < UU - 1) lab_lp[(size_t)bt * (UU - 1) + u] = xl - lse;
  }
}

// ---------------------------------------------------------------------------
// Kernel 2: E[b,t,u] = exclusive prefix sum over u of lab_lp, via
// V_WMMA_F32_16X16X4_F32 with staircase B matrices.
// One wave handles 16 (b,t) rows (M=16); 96 columns processed as 6 N-chunks
// of 16, each chunk = 4 chained WMMAs (K=4); running prefix carried through
// the accumulator (C <- broadcast of column 15 of D).
// ---------------------------------------------------------------------------
__global__ void k_cumsum_wmma(const float* __restrict__ lab_lp,  // [BT][96]
                              float* __restrict__ E) {           // [BT][97]
  const int gwave = (blockIdx.x * blockDim.x + threadIdx.x) >> 5;
  const int lane  = threadIdx.x & 31;
  const int tile  = gwave;                     // 0..127 (BT/16 tiles)
  if (tile >= BT / 16) return;

  const int half = lane >> 4;                  // 0: lanes 0-15, 1: lanes 16-31
  const int ln   = lane & 15;
  const int mrow = ln;                         // A-matrix M for this lane
  const int k0   = half * 2;                   // A/B K indices: k0, k0+1
  const int rowA = tile * 16 + mrow;

  // B_kc[k][n] = ((4*kc + k) <= n) ? 1 : 0   (4x16, staircase of ones)
  // B layout: lanes 0-15 hold K=0 (c0), K=1 (c1); lanes 16-31 hold K=2,3; N=ln
  v2f Bm[4];
#pragma unroll
  for (int kc = 0; kc < 4; ++kc) {
    Bm[kc].x = ((4 * kc + k0 + 0) <= ln) ? 1.f : 0.f;
    Bm[kc].y = ((4 * kc + k0 + 1) <= ln) ? 1.f : 0.f;
  }

  // E[row][0] = 0
  if (half == 0) E[(size_t)rowA * UU + 0] = 0.f;

  v8f acc = {};  // running carry lives in the accumulator
  const float* Arow = lab_lp + (size_t)rowA * (UU - 1);

#pragma unroll 1
  for (int c = 0; c < 6; ++c) {
    const int base = c * 16;
#pragma unroll
    for (int kc = 0; kc < 4; ++kc) {
      v2f A;
      const float* p = Arow + base + 4 * kc + k0;
      A.x = p[0];
      A.y = p[1];
      acc = __builtin_amdgcn_wmma_f32_16x16x4_f32(
          /*neg_a=*/false, A, /*neg_b=*/false, Bm[kc],
          /*c_mod=*/(short)0, acc, /*reuse_a=*/false, /*reuse_b=*/false);
    }
    // Store inclusive prefix S[base+ln] into E[row][base+ln+1].
    // D layout: VGPR r holds M=r (lanes 0-15) / M=r+8 (lanes 16-31), N=ln.
#pragma unroll
    for (int r = 0; r < 8; ++r) {
      const int M   = r + half * 8;
      const int row = tile * 16 + M;
      E[(size_t)row * UU + base + ln + 1] = acc[r];
    }
    // New carry C[m][*] = S at N=15 of row m.
#pragma unroll
    for (int r = 0; r < 8; ++r) {
      const float lo = __shfl(acc[r], 15, 32);  // carry for M=r
      const float hi = __shfl(acc[r], 31, 32);  // carry for M=r+8
      acc[r] = half ? hi : lo;
    }
  }
}

// ---------------------------------------------------------------------------
// Kernel 3: sequential alpha recurrence over t, cumulative logsumexp over u
// done as a wave32 Hillis-Steele scan in log space. One wave per batch.
// ---------------------------------------------------------------------------
__device__ __forceinline__ float lse2(float a, float b) {
  const float m = fmaxf(a, b);
  const float d = fminf(a, b) - m;             // <= 0 (NaN if both -inf)
  const float r = m + log1pf(__expf(d));
  return (m == NEG_INF) ? m : r;
}

__device__ __forceinline__ float scan_lse(float s, int lane) {
#pragma unroll
  for (int off = 1; off < 32; off <<= 1) {
    const float o = __shfl_up(s, (unsigned)off, 32);
    const float c = lse2(s, o);
    s = (lane >= off) ? c : s;
  }
  return s;
}

__global__ void k_alpha_scan(const float* __restrict__ blank_lp,  // [BT][97]
                             const float* __restrict__ E,         // [BT][97]
                             const int* __restrict__ f_len,
                             const int* __restrict__ y_len,
                             float* __restrict__ out) {
  const int b    = blockIdx.x;       // 0..7
  const int lane = threadIdx.x;      // 32 threads
  const int tcap = f_len[b] - 1;
  const int ucap = y_len[b];
  const size_t baseB = (size_t)b * TT;

  // alpha0 = E[b,0,:]
  const float* E0 = E + baseB * UU;
  float a0 = E0[lane];
  float a1 = E0[32 + lane];
  float a2 = E0[64 + lane];
  float a3 = (lane == 0) ? E0[96] : NEG_INF;

  const int cj = ucap >> 5, cl = ucap & 31;

  if (tcap == 0 && lane == cl) {
    const float av = (cj == 0) ? a0 : (cj == 1) ? a1 : (cj == 2) ? a2 : a3;
    out[b] = -(av + blank_lp[baseB * UU + ucap]);
  }

#pragma unroll 1
  for (int t = 1; t < TT; ++t) {
    const float* bl = blank_lp + (baseB + (t - 1)) * UU;
    const float* Et = E + (baseB + t) * UU;

    const float b0 = bl[lane], b1 = bl[32 + lane], b2 = bl[64 + lane];
    const float e0 = Et[lane], e1 = Et[32 + lane], e2 = Et[64 + lane];
    float b3 = 0.f, e3 = 0.f;
    if (lane == 0) { b3 = bl[96]; e3 = Et[96]; }

    // prefetch next iteration's rows (gfx1250 global_prefetch_b8)
    if (t + 1 < TT) {
      __builtin_prefetch(blank_lp + (baseB + t) * UU + lane * 4, 0, 0);
      __builtin_prefetch(E + (baseB + t + 1) * UU + lane * 4, 0, 0);
    }

    const float v0 = a0 + b0 - e0;
    const float v1 = a1 + b1 - e1;
    const float v2 = a2 + b2 - e2;
    const float v3 = (lane == 0) ? (a3 + b3 - e3) : NEG_INF;

    float s0 = scan_lse(v0, lane);
    float s1 = scan_lse(v1, lane);
    float s2 = scan_lse(v2, lane);
    const float c0 = __shfl(s0, 31, 32);
    s1 = lse2(s1, c0);
    const float c1 = __shfl(s1, 31, 32);
    s2 = lse2(s2, c1);
    const float c2 = __shfl(s2, 31, 32);
    const float s3 = lse2(v3, c2);

    a0 = e0 + s0;
    a1 = e1 + s1;
    a2 = e2 + s2;
    a3 = (lane == 0) ? (e3 + s3) : NEG_INF;

    if (t == tcap && lane == cl) {
      const float av = (cj == 0) ? a0 : (cj == 1) ? a1 : (cj == 2) ? a2 : a3;
      out[b] = -(av + blank_lp[(baseB + t) * UU + ucap]);
    }
  }
}

// ---------------------------------------------------------------------------
extern "C" void kernel_launch(void* const* d_in, const int* in_sizes, int n_in,
                              void* d_out, int out_size, void* d_ws, size_t ws_size,
                              hipStream_t stream) {
  const float* x      = (const float*)d_in[0];
  const int*   label  = (const int*)d_in[1];
  const int*   f_len  = (const int*)d_in[2];
  const int*   y_len  = (const int*)d_in[3];
  const int*   pblank = (const int*)d_in[4];
  float*       out    = (float*)d_out;

  float* blank_lp = (float*)d_ws;                 // BT*97 floats
  float* lab_lp   = blank_lp + (size_t)BT * UU;   // BT*96 floats
  float* Ebuf     = lab_lp + (size_t)BT * (UU-1); // BT*97 floats

  // Phase 1: streaming logsumexp + gathers (bandwidth bound, ~407MB read)
  dim3 g1(BT, (UU + 7) / 8);
  k_row_lse<<<g1, 256, 0, stream>>>(x, label, pblank, blank_lp, lab_lp);

  // Phase 2: exclusive cumsum over U via WMMA f32 16x16x4 (128 waves)
  k_cumsum_wmma<<<(BT / 16) / 8, 256, 0, stream>>>(lab_lp, Ebuf);

  // Phase 3: sequential alpha scan, one wave per batch
  k_alpha_scan<<<BB, 32, 0, stream>>>(blank_lp, Ebuf, f_len, y_len, out);
}